// Blur_249108103684
// MI455X (gfx1250) — compile-verified
//
#include <hip/hip_runtime.h>
#include <hip/hip_bf16.h>

typedef float v2f __attribute__((ext_vector_type(2)));
typedef float v4f __attribute__((ext_vector_type(4)));
typedef float v8f __attribute__((ext_vector_type(8)));

// 16-byte load with only 4-byte alignment guarantee (gfx1250 handles
// unaligned global b128; DWORD alignment mode requires only 4B).
// The empty asm tie keeps the i32 byte offset a per-iteration VGPR def so
// the backend's saddr+voffset (GlobalSAddr) pattern can match, instead of
// hoisting a 64-bit extended offset and building vaddr pairs per load.
__device__ __forceinline__ v4f load4u_s(const float* base, int byte_off) {
    asm("" : "+v"(byte_off));
    v4f v;
    __builtin_memcpy(&v, (const char*)base + byte_off, 16);
    return v;
}

__device__ __forceinline__ int iclamp(int v, int lo, int hi) {
    return v < lo ? lo : (v > hi ? hi : v);
}

// One wave = one 16x16 tile position, iterated over IMGS_PER_WAVE planes.
// Horizontal [1,3,3,1] pass in VALU -> T strip (per-lane column),
// vertical [1,3,3,1]/64 pass as 5 chained V_WMMA_F32_16X16X4_F32.
// Wave job id forced to SGPR via readfirstlane (uniform by construction):
// scalar plane bases, scalar branches, loop-invariant 32-bit byte-offset
// VGPRs for all loads/stores (saddr-form VMEM, clauses stay intact).
#define IMGS_PER_WAVE 8

__global__ __launch_bounds__(256) void blur4x4_wmma(
    const float* __restrict__ x, float* __restrict__ out,
    int n_jobs, int n_img)
{
    const int lane = threadIdx.x & 31;
    // threadIdx.x>>5 is identical for all 32 lanes of a wave: readfirstlane
    // is value-preserving here and moves the whole derivation to SGPRs.
    const int job = __builtin_amdgcn_readfirstlane(
        (int)(blockIdx.x * 8 + (threadIdx.x >> 5)));
    if (job >= n_jobs) return;          // scalar branch, whole wave exits

    const int tpos = job & 63;          // tile position in 8x8 grid
    const int grp  = job >> 6;          // image group
    const int th   = tpos >> 3;
    const int tw   = tpos & 7;
    const int h0   = th << 4;
    const int w0   = tw << 4;
    const int img0 = grp * IMGS_PER_WAVE;
    const int gcnt = min(IMGS_PER_WAVE, n_img - img0);

    const size_t plane = 128 * 128;
    const float* __restrict__ src = x   + (size_t)img0 * plane;
    float*       __restrict__ dst = out + (size_t)img0 * plane;

    const int nn   = lane & 15;         // N column (B/D) and M row (A) index
    const int off2 = (lane >> 4) << 1;  // 0 for lanes 0-15, 2 for lanes 16-31

    // ---- Loop-invariant setup (amortized over IMGS_PER_WAVE planes) ----

    // Vertical taps carry the full 1/64 normalization: kv = {1,3,3,1}/64
    const float kv0 = 1.0f / 64.0f, kv1 = 3.0f / 64.0f;

    // Banded A matrix, per ISA 16x4 f32 A layout:
    // lane holds A[i = nn, K = 4c + off2 + j], A[i,r] = kv[r-i] for r-i in [0,3]
    float a[5][2];
    #pragma unroll
    for (int c = 0; c < 5; ++c) {
        #pragma unroll
        for (int j = 0; j < 2; ++j) {
            const int u = (4 * c + off2 + j) - nn;
            float w = 0.0f;
            if (u == 0 || u == 3) w = kv0;
            if (u == 1 || u == 2) w = kv1;
            a[c][j] = w;
        }
    }

    // Clamped load BYTE offsets (valid for interior and edge tiles alike:
    // interior => clamps are identity), plus row-in-bounds masks.
    const int  cb  = w0 + nn - 2;
    const int  cbc = iclamp(cb, 0, 124);
    const int  dsh = cb - cbc;          // in [-2,1], fixed per lane
    int  roffB[5][2];
    bool hin  [5][2];
    #pragma unroll
    for (int c = 0; c < 5; ++c) {
        #pragma unroll
        for (int j = 0; j < 2; ++j) {
            const int r = 4 * c + off2 + j;
            const int h = h0 - 2 + r;
            roffB[c][j] = (iclamp(h, 0, 127) * 128 + cbc) * 4;
            hin  [c][j] = (h >= 0) & (h < 128);
        }
    }

    // Column-masked horizontal weights (edge path)
    float cw[4];
    #pragma unroll
    for (int tt = 0; tt < 4; ++tt) {
        const int col = cb + tt;
        const float w = (tt == 0 || tt == 3) ? 1.0f : 3.0f;
        cw[tt] = ((col >= 0) & (col < 128)) ? w : 0.0f;
    }

    // Scalar (SGPR) condition -> s_cbranch, EXEC stays all-ones for WMMA.
    const bool interior = (th >= 1) & (th <= 6) & (tw >= 1) & (tw <= 6);
    const int  sbase    = (h0 + ((lane >> 4) << 3)) * 128 + w0 + nn;

    // ---- Per-plane loop (scalar trip count, scalar base advance) ----
    for (int g = 0; g < gcnt; ++g) {
        float T[5][2];
        if (interior) {
            #pragma unroll
            for (int c = 0; c < 5; ++c) {
                #pragma unroll
                for (int j = 0; j < 2; ++j) {
                    const v4f v = load4u_s(src, roffB[c][j]);
                    T[c][j] = (v.x + v.w) + 3.0f * (v.y + v.z);
                }
            }
        } else {
            #pragma unroll
            for (int c = 0; c < 5; ++c) {
                #pragma unroll
                for (int j = 0; j < 2; ++j) {
                    const v4f v = load4u_s(src, roffB[c][j]);
                    float t = 0.0f;
                    #pragma unroll
                    for (int tt = 0; tt < 4; ++tt) {
                        const int   idx = tt + dsh; // in [0,3] when col in-bounds
                        const float val = (idx <= 0) ? v.x
                                        : (idx == 1) ? v.y
                                        : (idx == 2) ? v.z : v.w;
                        t = fmaf(cw[tt], val, t);
                    }
                    T[c][j] = hin[c][j] ? t : 0.0f;
                }
            }
        }

        // Vertical pass: out(16x16) = sum_c A_c(16x4) x T_c(4x16)
        v8f acc = {0.f, 0.f, 0.f, 0.f, 0.f, 0.f, 0.f, 0.f};
        #pragma unroll
        for (int c = 0; c < 5; ++c) {
            v2f av = {a[c][0], a[c][1]};
            v2f bv = {T[c][0], T[c][1]};
            acc = __builtin_amdgcn_wmma_f32_16x16x4_f32(
                /*neg_a=*/false, av, /*neg_b=*/false, bv,
                /*c_mod=*/(short)0, acc, /*reuse_a=*/false, /*reuse_b=*/false);
        }

        // D layout: lane holds column nn, rows m = vg + 8*(lane>=16);
        // vg*128 folds into the instruction immediate offset, x4 via
        // scale_offset on b32 stores.
        #pragma unroll
        for (int vg = 0; vg < 8; ++vg) {
            dst[sbase + vg * 128] = acc[vg];
        }

        src += plane;
        dst += plane;
    }
}

extern "C" void kernel_launch(void* const* d_in, const int* in_sizes, int n_in,
                              void* d_out, int out_size, void* d_ws, size_t ws_size,
                              hipStream_t stream) {
    (void)n_in; (void)out_size; (void)d_ws; (void)ws_size;
    const float* x   = (const float*)d_in[0];   // (16,256,128,128) fp32
    float*       out = (float*)d_out;           // same shape

    const int n_img  = in_sizes[0] / (128 * 128);             // 4096 planes
    const int n_grp  = (n_img + IMGS_PER_WAVE - 1) / IMGS_PER_WAVE;
    const int n_jobs = n_grp * 64;              // 8x8 tile positions per group
    const int blocks = (n_jobs + 7) / 8;        // 8 waves per block

    blur4x4_wmma<<<blocks, 256, 0, stream>>>(x, out, n_jobs, n_img);
}